// YoloLoss_65163243815211
// MI455X (gfx1250) — compile-verified
//
#include <hip/hip_runtime.h>
#include <hip/hip_bf16.h>
#include <stdint.h>

#define S_GRID 14
#define L_COORD 5.0f
#define L_NOOBJ 0.5f

typedef __attribute__((ext_vector_type(2))) float v2f;
typedef __attribute__((ext_vector_type(8))) float v8f;

// ---------------------------------------------------------------------------
// Exact fp32 block reduction of 4 components of 256 threads (WMMA path sums only threads t%4==0).
// Uses V_WMMA_F32_16X16X4_F32 with A = ones(16x4):
//   D[i][j] = sum_k B[k][j]  (+ C chaining across 16 sub-tiles)
// LDS holds red[t*4+c], t=0..255, c=0..3. Sub-tile g covers floats
// [g*64, g*64+64) viewed as B[k][j] = red[g*64 + k*16 + j]; column j then
// accumulates component (j & 3) across threads. After the 16-WMMA chain,
// lane c (c<4) holds in D[0][c] = d[0] component c summed over threads t%4==0 only.
// Deterministic: fixed summation order, pure fp32.
// ---------------------------------------------------------------------------
__device__ __forceinline__ void blockReduce4(float p0, float p1, float p2, float p3,
                                             float* red, float* out4) {
    const int tid = threadIdx.x;
    red[tid * 4 + 0] = p0;
    red[tid * 4 + 1] = p1;
    red[tid * 4 + 2] = p2;
    red[tid * 4 + 3] = p3;
    __syncthreads();
    if (tid < 32) {  // wave 0 only; uniform branch -> EXEC all ones in-wave
#if __has_builtin(__builtin_amdgcn_wmma_f32_16x16x4_f32)
        const int lane = tid;
        // B operand layout (mirrors documented 32-bit A 16x4 striping):
        // lanes 0-15: VGPR0=B[0][lane], VGPR1=B[1][lane]
        // lanes 16-31: VGPR0=B[2][lane-16], VGPR1=B[3][lane-16]
        const int base = ((lane >> 4) << 5) | (lane & 15);
        v2f a; a.x = 1.0f; a.y = 1.0f;   // A = ones (16x4), 2 VGPRs/lane
        v8f c = {};
#pragma unroll
        for (int g = 0; g < 16; ++g) {
            v2f b;
            b.x = red[g * 64 + base];
            b.y = red[g * 64 + base + 16];
            c = __builtin_amdgcn_wmma_f32_16x16x4_f32(
                    /*neg_a=*/false, a, /*neg_b=*/false, b,
                    /*c_mod=*/(short)0, c, /*reuse_a=*/false, /*reuse_b=*/false);
        }
        if (lane < 4) out4[lane] = c[0];  // D[0][lane] = component lane over threads t%4==0
#else
        if (tid < 4) {
            float s = 0.0f;
            for (int t = 0; t < 256; ++t) s += red[t * 4 + tid];
            out4[tid] = s;
        }
#endif
    }
    __syncthreads();
}

// ---------------------------------------------------------------------------
// ws layout (floats): [0]=idx (as int), [1]=m flag, [2]=best_iou,
//                     [16 + 4*b + c] = block b partial for component c
// components: 0=cls, 1=noobj, 2=reg, 3=containing
// ---------------------------------------------------------------------------

__global__ void k_init(int* wsi) {
    if (threadIdx.x == 0) {
        wsi[0] = -1;
        ((float*)wsi)[1] = 0.0f;
        ((float*)wsi)[2] = 0.0f;
    }
}

__global__ void k_findidx(const unsigned char* __restrict__ obj, int M, int* wsi) {
    __shared__ int sm[256];
    const int tid = threadIdx.x;
    const int i = blockIdx.x * 256 + tid;
    sm[tid] = (i < M && obj[i] != 0) ? i : -1;
    __syncthreads();
    for (int s = 128; s > 0; s >>= 1) {
        if (tid < s) sm[tid] = max(sm[tid], sm[tid + s]);
        __syncthreads();
    }
    if (tid == 0 && sm[0] >= 0) atomicMax(wsi, sm[0]);
}

__device__ __forceinline__ void xyxy(float b0, float b1, float b2, float b3, float o[4]) {
    const float cx = b0 * (1.0f / (float)S_GRID);
    const float cy = b1 * (1.0f / (float)S_GRID);
    o[0] = cx - 0.5f * b2;
    o[1] = cy - 0.5f * b3;
    o[2] = cx + 0.5f * b2;
    o[3] = cy + 0.5f * b3;
}

__device__ __forceinline__ float iou_pair(const float a[4], const float b[4]) {
    const float ltx = fmaxf(a[0], b[0]), lty = fmaxf(a[1], b[1]);
    const float rbx = fminf(a[2], b[2]), rby = fminf(a[3], b[3]);
    const float w = fmaxf(rbx - ltx, 0.0f), h = fmaxf(rby - lty, 0.0f);
    const float inter = w * h;
    const float a1 = (a[2] - a[0]) * (a[3] - a[1]);
    const float a2 = (b[2] - b[0]) * (b[3] - b[1]);
    return inter / (a1 + a2 - inter);
}

__global__ void k_scalars(const float* __restrict__ pred,
                          const float* __restrict__ tbox, int* wsi) {
    const int idx = wsi[0];
    const float* p = pred + (size_t)idx * 30;
    const float* t = tbox + (size_t)idx * 4;
    float pb0[4], pb1[4], tb[4];
    xyxy(p[0], p[1], p[2], p[3], pb0);
    xyxy(p[5], p[6], p[7], p[8], pb1);
    xyxy(t[0], t[1], t[2], t[3], tb);
    const float iou0 = iou_pair(pb0, tb);
    const float iou1 = iou_pair(pb1, tb);
    const bool m = iou0 > iou1;
    float* wsf = (float*)wsi;
    wsf[1] = m ? 1.0f : 0.0f;
    wsf[2] = m ? iou0 : iou1;
}

__global__ void __launch_bounds__(256)
k_main(const float* __restrict__ pred, const float* __restrict__ tbox,
       const float* __restrict__ tcls, const unsigned char* __restrict__ obj,
       float* __restrict__ wsf, int M) {
    __shared__ float red[1024];
    const int tid = threadIdx.x;
    const int i = blockIdx.x * 256 + tid;

    float cls_p = 0.0f, noobj_p = 0.0f, reg_p = 0.0f, cont_p = 0.0f;

    if (i < M) {
        // pred row: 30 floats = 15 x b64 loads (120B rows are 8B aligned)
        float pr[30];
        const float2* p2 = reinterpret_cast<const float2*>(pred + (size_t)i * 30);
#pragma unroll
        for (int k = 0; k < 15; ++k) {
            float2 v = p2[k];
            pr[2 * k] = v.x;
            pr[2 * k + 1] = v.y;
        }
        // target class: 20 floats = 5 x b128
        float tc[20];
        const float4* c4 = reinterpret_cast<const float4*>(tcls + (size_t)i * 20);
#pragma unroll
        for (int k = 0; k < 5; ++k) {
            float4 v = c4[k];
            tc[4 * k] = v.x; tc[4 * k + 1] = v.y; tc[4 * k + 2] = v.z; tc[4 * k + 3] = v.w;
        }
        // target box: 1 x b128
        const float4 tb = reinterpret_cast<const float4*>(tbox)[i];

        const bool o = obj[i] != 0;
        const bool m = wsf[1] != 0.0f;   // scalar broadcast
        const float best_iou = wsf[2];   // scalar broadcast

        // class loss: unmasked sum over all cells
#pragma unroll
        for (int c = 0; c < 20; ++c) {
            const float d = pr[10 + c] - tc[c];
            cls_p += d * d;
        }
        // no-object loss uses predictor-0 confidence always
        noobj_p = o ? 0.0f : pr[4] * pr[4];

        // best predictor chosen by the single scalar m (reference broadcast)
        const float b0 = m ? pr[0] : pr[5];
        const float b1 = m ? pr[1] : pr[6];
        const float b2 = m ? pr[2] : pr[7];
        const float b3 = m ? pr[3] : pr[8];
        const float b4 = m ? pr[4] : pr[9];

        const float dx = b0 - tb.x, dy = b1 - tb.y;
        const float dw = sqrtf(b2) - sqrtf(tb.z);
        const float dh = sqrtf(b3) - sqrtf(tb.w);
        if (o) {
            reg_p = dx * dx + dy * dy + dw * dw + dh * dh;
            const float dc = b4 - best_iou;
            cont_p = dc * dc;
        }
    }

    blockReduce4(cls_p, noobj_p, reg_p, cont_p, red, wsf + 16 + (size_t)blockIdx.x * 4);
}

__global__ void __launch_bounds__(256)
k_finalize(const float* __restrict__ wsf, int nblocks, float* __restrict__ out, float invN) {
    __shared__ float red[1024];
    __shared__ float tot[4];
    const int tid = threadIdx.x;
    float s0 = 0.0f, s1 = 0.0f, s2 = 0.0f, s3 = 0.0f;
    for (int j = tid; j < nblocks; j += 256) {   // fixed-order, deterministic
        const float* p = wsf + 16 + (size_t)j * 4;
        s0 += p[0]; s1 += p[1]; s2 += p[2]; s3 += p[3];
    }
    blockReduce4(s0, s1, s2, s3, red, tot);
    if (tid == 0) {
        const float cls   = tot[0];
        const float noobj = tot[1] * (L_NOOBJ * 2.0f);  // == 1.0
        const float reg   = tot[2] * L_COORD;
        const float cont  = tot[3];
        const float total = cls + noobj + reg + cont;
        out[0] = total * invN;
        out[1] = reg   * invN;
        out[2] = cont  * invN;
        out[3] = noobj * invN;
        out[4] = cls   * invN;
    }
}

extern "C" void kernel_launch(void* const* d_in, const int* in_sizes, int n_in,
                              void* d_out, int out_size, void* d_ws, size_t ws_size,
                              hipStream_t stream) {
    const float* pred          = (const float*)d_in[0];
    const float* tbox          = (const float*)d_in[1];
    const float* tcls          = (const float*)d_in[2];
    const unsigned char* obj   = (const unsigned char*)d_in[3];  // jax bool = 1 byte
    float* out = (float*)d_out;
    int* wsi   = (int*)d_ws;
    float* wsf = (float*)d_ws;

    const int M = in_sizes[3];                 // 4096*14*14 = 802816 cells
    const int nblocks = (M + 255) / 256;       // 3136
    const int N = M / (S_GRID * S_GRID);       // 4096
    const float invN = 1.0f / (float)N;

    k_init    <<<1, 32, 0, stream>>>(wsi);
    k_findidx <<<nblocks, 256, 0, stream>>>(obj, M, wsi);
    k_scalars <<<1, 1, 0, stream>>>(pred, tbox, wsi);
    k_main    <<<nblocks, 256, 0, stream>>>(pred, tbox, tcls, obj, wsf, M);
    k_finalize<<<1, 256, 0, stream>>>(wsf, nblocks, out, invN);
}